// NNSubmulti_21784074125886
// MI455X (gfx1250) — compile-verified
//
#include <hip/hip_runtime.h>
#include <hip/hip_bf16.h>

// ---------------------------------------------------------------------------
// CDNA5 WMMA f32 16x16x4 fragments (wave32):
//   A(16x4)  = 2 f32/lane, B(4x16) = 2 f32/lane, C/D(16x16) = 8 f32/lane
// ---------------------------------------------------------------------------
typedef __attribute__((ext_vector_type(2))) float v2f;
typedef __attribute__((ext_vector_type(8))) float v8f;

#define BATCH 16
#define TT    128   // TQ == TK == 128
#define EMB   256
#define HH    100
#define HPAD  112   // 7 tiles of 16

__device__ __forceinline__ v8f wmma_f32_k4(v2f a, v2f b, v8f c) {
  // (neg_a, A, neg_b, B, c_mod, C, reuse_a, reuse_b)
  return __builtin_amdgcn_wmma_f32_16x16x4_f32(false, a, false, b, (short)0, c,
                                               false, false);
}

__device__ __forceinline__ float fast_tanh(float x) {
#if __has_builtin(__builtin_amdgcn_tanhf)
  return __builtin_amdgcn_tanhf(x);
#else
  float r;
  asm volatile("v_tanh_f32 %0, %1" : "=v"(r) : "v"(x));
  return r;
#endif
}

// ---------------------------------------------------------------------------
// Weight packing: k-pair interleaved, zero-padded, k-major:
//   BP[(k>>1)*N + n][k&1]  -> one lane's B fragment {k,k+1} is ONE b64 load.
// ---------------------------------------------------------------------------
__global__ void pack_w1(const float* __restrict__ W1,
                        float* __restrict__ W1aP, float* __restrict__ W1bP) {
  const int k = blockIdx.x;    // 0..255
  const int n = threadIdx.x;   // 0..111
  const float va = (n < HH) ? W1[n * (2 * EMB) + k] : 0.0f;
  const float vb = (n < HH) ? W1[n * (2 * EMB) + EMB + k] : 0.0f;
  const int idx = ((k >> 1) * HPAD + n) * 2 + (k & 1);
  W1aP[idx] = va;
  W1bP[idx] = vb;
}

__global__ void pack_wlast(const float* __restrict__ Wlast,
                           float* __restrict__ WlastP) {
  const int k = blockIdx.x;    // 0..511
  const int n = threadIdx.x;   // 0..255
  WlastP[((k >> 1) * EMB + n) * 2 + (k & 1)] = Wlast[n * (2 * EMB) + k];
}

// qP[b][(k>>1)*EMB + n][k&1] = queries[b][k][n]   (for keys_attn = P @ q)
__global__ void pack_q(const float* __restrict__ q, float* __restrict__ qP) {
  const int b = blockIdx.y;
  const int k = blockIdx.x;    // 0..127
  const int n = threadIdx.x;   // 0..255
  qP[(((b * (TT / 2) + (k >> 1)) * EMB) + n) * 2 + (k & 1)] =
      q[(b * TT + k) * EMB + n];
}

// ---------------------------------------------------------------------------
// Batched f32 WMMA GEMM:  D[b] = A[b](MxK row-major) * B[b](KxN), with B
// pre-packed pair-interleaved (v2f per (k-pair, n)).  One wave per
// (16 x 16*NT) strip; A fragment reused across NT WMMAs; double-buffered
// K-loop so loads for step k+4 overlap the WMMAs of step k.
// Fragment layout (32-bit, wave32):
//   A:  lane L -> row  M = L&15, cols {k0+2*(L>>4), +1}
//   B:  lane L -> col  N = L&15, rows {k0+2*(L>>4), +1}
//   C/D lane L, vgpr r -> row M = r + 8*(L>>4), col N = L&15
// ---------------------------------------------------------------------------
template <int NT>
__global__ void gemm_wmma_f32(const float* __restrict__ A, int a_batch, int lda,
                              const float* __restrict__ BP, int b_batch,
                              int nld, float* __restrict__ D, int d_batch,
                              int ldd, int transD, int K) {
  const int bb = blockIdx.z;
  const int m0 = blockIdx.x * 16;
  const int n0 = blockIdx.y * (16 * NT);

  const int lane = threadIdx.x;
  const int half = lane >> 4;       // 0: K pair {0,1}, 1: K pair {2,3}
  const int lm   = lane & 15;
  const int m    = m0 + lm;

  const float* arow = A + bb * a_batch + m * lda + 2 * half;  // k-pair base
  const v2f*   bv   = (const v2f*)(BP + bb * b_batch) + half * nld + n0 + lm;

  v8f acc[NT];
#pragma unroll
  for (int t = 0; t < NT; ++t) acc[t] = (v8f){};

  // prologue: fragments for k0 = 0
  v2f afc = *(const v2f*)arow;
  v2f bfc[NT];
#pragma unroll
  for (int t = 0; t < NT; ++t) bfc[t] = bv[t * 16];

#pragma unroll 2
  for (int k0 = 0; k0 < K - 4; k0 += 4) {
    // prefetch k0+4 while the WMMAs below execute
    const v2f* bn = bv + ((k0 + 4) >> 1) * nld;
    v2f afn = *(const v2f*)(arow + k0 + 4);
    v2f bfn[NT];
#pragma unroll
    for (int t = 0; t < NT; ++t) bfn[t] = bn[t * 16];

#pragma unroll
    for (int t = 0; t < NT; ++t) acc[t] = wmma_f32_k4(afc, bfc[t], acc[t]);

    afc = afn;
#pragma unroll
    for (int t = 0; t < NT; ++t) bfc[t] = bfn[t];
  }
  // epilogue k-step
#pragma unroll
  for (int t = 0; t < NT; ++t) acc[t] = wmma_f32_k4(afc, bfc[t], acc[t]);

  float* Db = D + bb * d_batch;
#pragma unroll
  for (int t = 0; t < NT; ++t) {
    const int n = n0 + t * 16 + lm;
#pragma unroll
    for (int r = 0; r < 8; ++r) {
      const int mr = m0 + r + 8 * half;
      if (transD) Db[n * ldd + mr] = acc[t][r];
      else        Db[mr * ldd + n] = acc[t][r];
    }
  }
}

// ---------------------------------------------------------------------------
// Score + softmax.  One 128-thread block per (b, i): thread j accumulates
// sim = sum_h W2[h]*tanh(a[b,i,h]+b1[h] + cT[b,h,j]); cT layout makes the
// inner read coalesced.  Hardware v_tanh_f32, block softmax over j.
// ---------------------------------------------------------------------------
__global__ void score_softmax(const float* __restrict__ a,
                              const float* __restrict__ cT,
                              const float* __restrict__ b1,
                              const float* __restrict__ W2,
                              const float* __restrict__ qmask,
                              const float* __restrict__ kmask,
                              float* __restrict__ P) {
  const int b = blockIdx.y;
  const int i = blockIdx.x;
  const int j = threadIdx.x;  // 0..127

  __shared__ float a_sh[HH];
  __shared__ float w2_sh[HH];
  __shared__ float red[TT];

  if (j < HH) {
    a_sh[j]  = a[(b * TT + i) * HPAD + j] + b1[j];  // fold bias into a-row
    w2_sh[j] = W2[j];
  }
  __syncthreads();

  const float* c = cT + b * HPAD * TT;
  float s = 0.0f;
#pragma unroll 4
  for (int h = 0; h < HH; ++h) {
    const float x = a_sh[h] + c[h * TT + j];   // coalesced over j
    s = fmaf(w2_sh[h], fast_tanh(x), s);
  }

  const float NEG = -4294967295.0f;            // -(2^32) + 1
  float sim = (qmask[b * TT + j] == 0.0f) ? NEG : s;

  red[j] = sim; __syncthreads();
#pragma unroll
  for (int off = 64; off > 0; off >>= 1) {
    if (j < off) red[j] = fmaxf(red[j], red[j + off]);
    __syncthreads();
  }
  const float mx = red[0];
  __syncthreads();

  const float e = __expf(sim - mx);
  red[j] = e; __syncthreads();
#pragma unroll
  for (int off = 64; off > 0; off >>= 1) {
    if (j < off) red[j] += red[j + off];
    __syncthreads();
  }
  const float p = e * (1.0f / red[0]) * kmask[b * TT + i];
  P[(b * TT + i) * TT + j] = p;
}

// ---------------------------------------------------------------------------
// Final: out = relu([ka*keys , (ka-keys)^2] @ Wlast^T + blast)
// Feature fabricated on the fly into the A fragment (b64 pair loads);
// B from packed WlastP.  NT n-tiles per wave, double-buffered, K = 512.
// ---------------------------------------------------------------------------
template <int NT>
__global__ void final_wmma(const float* __restrict__ ka,
                           const float* __restrict__ keys,
                           const float* __restrict__ WlastP,
                           const float* __restrict__ blast,
                           float* __restrict__ out) {
  const int bb = blockIdx.z;
  const int m0 = blockIdx.x * 16;
  const int n0 = blockIdx.y * (16 * NT);

  const int lane = threadIdx.x;
  const int half = lane >> 4;
  const int lm   = lane & 15;
  const int m    = m0 + lm;

  const float* kab = ka   + (bb * TT + m) * EMB;
  const float* kb  = keys + (bb * TT + m) * EMB;
  const v2f*   bv  = (const v2f*)WlastP + half * EMB + n0 + lm;

  v8f acc[NT];
#pragma unroll
  for (int t = 0; t < NT; ++t) acc[t] = (v8f){};

  auto make_a = [&](int kk) -> v2f {
    const int e = kk & (EMB - 1);
    const v2f x = *(const v2f*)(kab + e);
    const v2f y = *(const v2f*)(kb + e);
    v2f r;
    if (kk < EMB) {
      r.x = x.x * y.x;
      r.y = x.y * y.y;
    } else {
      const float d0 = x.x - y.x, d1 = x.y - y.y;
      r.x = d0 * d0;
      r.y = d1 * d1;
    }
    return r;
  };

  v2f afc = make_a(2 * half);
  v2f bfc[NT];
#pragma unroll
  for (int t = 0; t < NT; ++t) bfc[t] = bv[t * 16];

#pragma unroll 2
  for (int k0 = 0; k0 < 2 * EMB - 4; k0 += 4) {
    const v2f* bn = bv + ((k0 + 4) >> 1) * EMB;
    v2f afn = make_a(k0 + 4 + 2 * half);
    v2f bfn[NT];
#pragma unroll
    for (int t = 0; t < NT; ++t) bfn[t] = bn[t * 16];

#pragma unroll
    for (int t = 0; t < NT; ++t) acc[t] = wmma_f32_k4(afc, bfc[t], acc[t]);

    afc = afn;
#pragma unroll
    for (int t = 0; t < NT; ++t) bfc[t] = bfn[t];
  }
#pragma unroll
  for (int t = 0; t < NT; ++t) acc[t] = wmma_f32_k4(afc, bfc[t], acc[t]);

#pragma unroll
  for (int t = 0; t < NT; ++t) {
    const int n = n0 + t * 16 + lm;
    const float bvb = blast[n];
#pragma unroll
    for (int r = 0; r < 8; ++r) {
      const int mr = m0 + r + 8 * half;
      float v = acc[t][r] + bvb;
      v = v > 0.0f ? v : 0.0f;
      out[bb * TT * EMB + mr * EMB + n] = v;
    }
  }
}

// ---------------------------------------------------------------------------
extern "C" void kernel_launch(void* const* d_in, const int* in_sizes, int n_in,
                              void* d_out, int out_size, void* d_ws,
                              size_t ws_size, hipStream_t stream) {
  const float* queries = (const float*)d_in[0];  // [16,128,256]
  const float* keys    = (const float*)d_in[1];  // [16,128,256]
  const float* qmask   = (const float*)d_in[2];  // [16,128]
  const float* kmask   = (const float*)d_in[3];  // [16,128]
  const float* W1      = (const float*)d_in[4];  // [100,512]
  const float* b1      = (const float*)d_in[5];  // [100]
  const float* W2      = (const float*)d_in[6];  // [1,100]
  const float* Wlast   = (const float*)d_in[7];  // [256,512]
  const float* blast   = (const float*)d_in[8];  // [256]
  float* out = (float*)d_out;                    // [16,128,256]

  float* ws      = (float*)d_ws;
  float* a_ws    = ws;                             // [16][128][112]
  float* cT_ws   = a_ws    + BATCH * TT * HPAD;    // [16][112][128]
  float* P_ws    = cT_ws   + BATCH * HPAD * TT;    // [16][128][128]
  float* ka_ws   = P_ws    + BATCH * TT * TT;      // [16][128][256]
  float* W1aP_ws = ka_ws   + BATCH * TT * EMB;     // [128][112][2]
  float* W1bP_ws = W1aP_ws + EMB * HPAD;           // [128][112][2]
  float* WlP_ws  = W1bP_ws + EMB * HPAD;           // [256][256][2]
  float* qP_ws   = WlP_ws  + 2 * EMB * EMB;        // [16][64][256][2]

  // 0) pack weights + queries (pair-interleaved k-major, zero-padded)
  pack_w1<<<dim3(EMB), HPAD, 0, stream>>>(W1, W1aP_ws, W1bP_ws);
  pack_wlast<<<dim3(2 * EMB), EMB, 0, stream>>>(Wlast, WlP_ws);
  pack_q<<<dim3(TT, BATCH), EMB, 0, stream>>>(queries, qP_ws);

  // 1) a  = keys @ W1a^T        -> [b][i][h]   (NT=7 covers all 112 cols)
  gemm_wmma_f32<7><<<dim3(TT / 16, 1, BATCH), 32, 0, stream>>>(
      keys, TT * EMB, EMB, W1aP_ws, 0, HPAD,
      a_ws, TT * HPAD, HPAD, /*transD=*/0, EMB);

  // 2) cT = (queries @ W1b^T)^T -> [b][h][j]   (coalesced for scoring)
  gemm_wmma_f32<7><<<dim3(TT / 16, 1, BATCH), 32, 0, stream>>>(
      queries, TT * EMB, EMB, W1bP_ws, 0, HPAD,
      cT_ws, HPAD * TT, TT, /*transD=*/1, EMB);

  // 3) sim -> softmax -> P      (hardware tanh, block softmax)
  score_softmax<<<dim3(TT, BATCH), TT, 0, stream>>>(
      a_ws, cT_ws, b1, W2, qmask, kmask, P_ws);

  // 4) keys_attn = P @ queries  (packed qP, K=128)
  gemm_wmma_f32<4><<<dim3(TT / 16, EMB / 64, BATCH), 32, 0, stream>>>(
      P_ws, TT * TT, TT, qP_ws, (TT / 2) * EMB * 2, EMB,
      ka_ws, TT * EMB, EMB, /*transD=*/0, TT);

  // 5) out = relu(feature @ Wlast^T + blast), feature fused into A fragment
  final_wmma<4><<<dim3(TT / 16, EMB / 64, BATCH), 32, 0, stream>>>(
      ka_ws, keys, WlP_ws, blast, out);
}